// AutoRegMLAttention_83339545412133
// MI455X (gfx1250) — compile-verified
//
#include <hip/hip_runtime.h>
#include <hip/hip_bf16.h>

#define B_   4
#define T_   4096
#define Tp1  4097
#define HID  4096
#define CQ_  1536
#define CKV_ 512
#define NH_  32
#define HD_  128
#define RD_  64

#define INV_SCALE 0.0721687836f   /* 1/sqrt(HD+RD) = 1/sqrt(192) */

typedef float v2f __attribute__((ext_vector_type(2)));
typedef float v8f __attribute__((ext_vector_type(8)));

__device__ __forceinline__ v8f wmma_f32_4(v2f a, v2f b, v8f c) {
  return __builtin_amdgcn_wmma_f32_16x16x4_f32(false, a, false, b, (short)0, c, false, false);
}

// ---------------------------------------------------------------------------
// Generic batched GEMV: y[b][j] = sum_i x[b][i] * W[i][j]   (W row-major K x N)
// grid = (ceil(N/256), B), block = 256, dynamic LDS = K floats
// ---------------------------------------------------------------------------
__global__ __launch_bounds__(256) void gemv_k(const float* __restrict__ x,
                                              const float* __restrict__ W,
                                              float* __restrict__ y,
                                              int K, int N) {
  extern __shared__ float xs[];
  const int b = blockIdx.y;
  for (int i = threadIdx.x; i < K; i += 256) xs[i] = x[(size_t)b * K + i];
  __syncthreads();
  const int j = blockIdx.x * 256 + threadIdx.x;
  if (j >= N) return;
  float acc = 0.f;
  const float* __restrict__ wp = W + j;
#pragma unroll 4
  for (int i = 0; i < K; ++i) acc = fmaf(xs[i], wp[(size_t)i * N], acc);
  y[(size_t)b * N + j] = acc;
}

// ---------------------------------------------------------------------------
// q_abs[b][n][v] = sum_d qC[b][n][d] * W_UK_C[v][n][d]
// grid = (NH, B), block = 256
// ---------------------------------------------------------------------------
__global__ __launch_bounds__(256) void qabs_k(const float* __restrict__ qC,
                                              const float* __restrict__ Wuk,
                                              float* __restrict__ q_abs) {
  __shared__ float qh[HD_];
  const int b = blockIdx.y, n = blockIdx.x;
  if (threadIdx.x < HD_) qh[threadIdx.x] = qC[((size_t)b * NH_ + n) * HD_ + threadIdx.x];
  __syncthreads();
  for (int v = threadIdx.x; v < CKV_; v += 256) {
    const float* __restrict__ w = Wuk + ((size_t)v * NH_ + n) * HD_;
    float acc = 0.f;
#pragma unroll 8
    for (int d = 0; d < HD_; ++d) acc = fmaf(qh[d], w[d], acc);
    q_abs[((size_t)b * NH_ + n) * CKV_ + v] = acc;
  }
}

// ---------------------------------------------------------------------------
// Prefix content scores via WMMA:
//   scores[b][n][t] = sum_v cached_cKV[b][t][v] * q_abs[b][n][v]   (raw, unscaled)
// grid = (T/128, B), block = 256 (8 waves, each wave one 16-row t tile,
// both head groups). A = cKV 16x4 tile, B = q_abs (staged in LDS in B layout).
// ---------------------------------------------------------------------------
__global__ __launch_bounds__(256) void scores_wmma_k(const float* __restrict__ ckv,
                                                     const float* __restrict__ q_abs,
                                                     float* __restrict__ scores) {
  // qs layout: element (k, n) at qs[(k>>1)*64 + n*2 + (k&1)]  -> v2f per lane
  __shared__ float qs[CKV_ * NH_];  // 16384 floats = 64 KB
  const int b = blockIdx.y;
  for (int i = threadIdx.x; i < NH_ * CKV_; i += 256) {
    const int n = i >> 9, k = i & (CKV_ - 1);
    qs[(k >> 1) * (2 * NH_) + n * 2 + (k & 1)] = q_abs[((size_t)b * NH_ + n) * CKV_ + k];
  }
  __syncthreads();

  const int wave = threadIdx.x >> 5;
  const int lane = threadIdx.x & 31;
  const int t0   = (blockIdx.x * 8 + wave) * 16;
  const int m    = lane & 15;            // A row (t within tile) / B column (head)
  const int koff = (lane >> 4) * 2;      // which K pair this half-wave holds

  const float* __restrict__ arow = ckv + ((size_t)b * T_ + t0 + m) * CKV_;
  v8f acc0 = {}, acc1 = {};
  for (int k = 0; k < CKV_; k += 4) {
    v2f a = *(const v2f*)(arow + k + koff);
    const float* qp = qs + ((k + koff) >> 1) * (2 * NH_);
    v2f b0 = *(const v2f*)(qp + m * 2);
    v2f b1 = *(const v2f*)(qp + (m + 16) * 2);
    acc0 = wmma_f32_4(a, b0, acc0);
    acc1 = wmma_f32_4(a, b1, acc1);
  }
  // D layout: lanes 0-15 -> M=r, N=lane; lanes 16-31 -> M=8+r, N=lane-16
  const int tg = (lane >> 4) * 8;
  float* p0 = scores + ((size_t)(b * NH_ + m) * Tp1) + t0 + tg;
  float* p1 = scores + ((size_t)(b * NH_ + m + 16) * Tp1) + t0 + tg;
#pragma unroll
  for (int r = 0; r < 8; ++r) { p0[r] = acc0[r]; p1[r] = acc1[r]; }
}

// ---------------------------------------------------------------------------
// Add rope scores (qR . cached_kR), scale, add mask. Covers t < T.
// grid = (T/8, B), block = 256  (8 t-rows x 32 heads per block)
// ---------------------------------------------------------------------------
__global__ __launch_bounds__(256) void rope_score_k(const float* __restrict__ kR,
                                                    const float* __restrict__ qR,
                                                    const float* __restrict__ mask,
                                                    float* __restrict__ scores) {
  __shared__ float qRl[NH_ * RD_];  // 8 KB
  const int b = blockIdx.y;
  for (int i = threadIdx.x; i < NH_ * RD_; i += 256) qRl[i] = qR[(size_t)b * NH_ * RD_ + i];
  __syncthreads();
  const int n = threadIdx.x & 31;
  const int t = blockIdx.x * 8 + (threadIdx.x >> 5);
  const float4* __restrict__ kp =
      (const float4*)(kR + (((size_t)b * T_ + t) * NH_ + n) * RD_);
  __builtin_prefetch(kp + 8 * NH_ * RD_ / 4, 0, 0);  // next block's rows
  const float4* __restrict__ qp = (const float4*)(qRl + n * RD_);
  float acc = 0.f;
#pragma unroll
  for (int r = 0; r < RD_ / 4; ++r) {
    float4 kv = kp[r], qv = qp[r];
    acc = fmaf(kv.x, qv.x, acc);
    acc = fmaf(kv.y, qv.y, acc);
    acc = fmaf(kv.z, qv.z, acc);
    acc = fmaf(kv.w, qv.w, acc);
  }
  const size_t si = (size_t)(b * NH_ + n) * Tp1 + t;
  scores[si] = (scores[si] + acc) * INV_SCALE + mask[si] * (-1e9f);
}

// ---------------------------------------------------------------------------
// Score for the newly generated token (index T): qC.kC + qR.kR_t
// launch <<<1,128>>>
// ---------------------------------------------------------------------------
__global__ void new_score_k(const float* __restrict__ qC, const float* __restrict__ kC,
                            const float* __restrict__ qR, const float* __restrict__ kRt,
                            const float* __restrict__ mask, float* __restrict__ scores) {
  const int b = threadIdx.x >> 5, n = threadIdx.x & 31;
  const size_t hn = (size_t)b * NH_ + n;
  const float* qc = qC + hn * HD_;
  const float* kc = kC + hn * HD_;
  const float* qr = qR + hn * RD_;
  const float* kr = kRt + hn * RD_;
  float acc = 0.f;
  for (int d = 0; d < HD_; ++d) acc = fmaf(qc[d], kc[d], acc);
  for (int r = 0; r < RD_; ++r) acc = fmaf(qr[r], kr[r], acc);
  const size_t si = hn * Tp1 + T_;
  scores[si] = acc * INV_SCALE + mask[si] * (-1e9f);
}

// ---------------------------------------------------------------------------
// Softmax over T+1 per (b, n). grid = (NH, B), block = 256. In-place.
// ---------------------------------------------------------------------------
__global__ __launch_bounds__(256) void softmax_k(float* __restrict__ scores) {
  __shared__ float red[256];
  const int b = blockIdx.y, n = blockIdx.x;
  float* s = scores + (size_t)(b * NH_ + n) * Tp1;
  float mx = -1e30f;
  for (int i = threadIdx.x; i < Tp1; i += 256) mx = fmaxf(mx, s[i]);
  red[threadIdx.x] = mx;
  __syncthreads();
  for (int o = 128; o > 0; o >>= 1) {
    if (threadIdx.x < o) red[threadIdx.x] = fmaxf(red[threadIdx.x], red[threadIdx.x + o]);
    __syncthreads();
  }
  mx = red[0];
  __syncthreads();
  float sum = 0.f;
  for (int i = threadIdx.x; i < Tp1; i += 256) {
    const float e = __expf(s[i] - mx);
    s[i] = e;
    sum += e;
  }
  red[threadIdx.x] = sum;
  __syncthreads();
  for (int o = 128; o > 0; o >>= 1) {
    if (threadIdx.x < o) red[threadIdx.x] += red[threadIdx.x + o];
    __syncthreads();
  }
  const float inv = 1.f / red[0];
  for (int i = threadIdx.x; i < Tp1; i += 256) s[i] *= inv;
}

// ---------------------------------------------------------------------------
// Aggregation via WMMA (absorbed V):
//   ctx[b][n][v] = sum_{t<T} probs[b][n][t]*cKV[b][t][v] + probs[b][n][T]*cKV_t[b][v]
// grid = (CKV/128, B), block = 256: 8 waves, each wave one 16-col v tile,
// both head groups (M = 32 heads via two accumulators sharing the B operand).
// ---------------------------------------------------------------------------
__global__ __launch_bounds__(256) void agg_wmma_k(const float* __restrict__ probs,
                                                  const float* __restrict__ ckv,
                                                  const float* __restrict__ ckv_new,
                                                  float* __restrict__ ctx) {
  // pA layout: element (tt, h) at pA[(tt>>1)*64 + h*2 + (tt&1)], tt in [0,128)
  __shared__ float pA[64 * 64];    // 16 KB
  __shared__ float ptail[NH_];
  const int b = blockIdx.y;
  if (threadIdx.x < NH_)
    ptail[threadIdx.x] = probs[(size_t)(b * NH_ + threadIdx.x) * Tp1 + T_];

  const int wave = threadIdx.x >> 5;
  const int lane = threadIdx.x & 31;
  const int m    = lane & 15;
  const int koff = (lane >> 4) * 2;
  const int v    = (blockIdx.x * 8 + wave) * 16 + m;

  v8f acc0 = {}, acc1 = {};
  for (int tc = 0; tc < T_; tc += 128) {
    __syncthreads();
    for (int i = threadIdx.x; i < NH_ * 128; i += 256) {
      const int h = i >> 7, tt = i & 127;
      pA[(tt >> 1) * 64 + h * 2 + (tt & 1)] = probs[(size_t)(b * NH_ + h) * Tp1 + tc + tt];
    }
    __syncthreads();
    for (int kk = 0; kk < 128; kk += 4) {
      const float* brow = ckv + ((size_t)b * T_ + tc + kk + koff) * CKV_ + v;
      v2f bm = { brow[0], brow[CKV_] };
      const float* ap = pA + ((kk + koff) >> 1) * 64;
      v2f a0 = *(const v2f*)(ap + m * 2);
      v2f a1 = *(const v2f*)(ap + (m + 16) * 2);
      acc0 = wmma_f32_4(a0, bm, acc0);
      acc1 = wmma_f32_4(a1, bm, acc1);
    }
  }
  // tail: t = T row comes from the fresh cKV_t
  const float cv = ckv_new[(size_t)b * CKV_ + v];
  const int hh = (lane >> 4) * 8;
#pragma unroll
  for (int r = 0; r < 8; ++r) {
    acc0[r] = fmaf(ptail[hh + r], cv, acc0[r]);
    acc1[r] = fmaf(ptail[16 + hh + r], cv, acc1[r]);
  }
#pragma unroll
  for (int r = 0; r < 8; ++r) {
    ctx[(size_t)(b * NH_ + hh + r) * CKV_ + v] = acc0[r];
    ctx[(size_t)(b * NH_ + 16 + hh + r) * CKV_ + v] = acc1[r];
  }
}

// ---------------------------------------------------------------------------
// attn_out[b][n][d] = sum_v ctx[b][n][v] * W_UV_C[v][n][d]
// grid = (NH, B), block = 128
// ---------------------------------------------------------------------------
__global__ __launch_bounds__(128) void attnout_k(const float* __restrict__ ctx,
                                                 const float* __restrict__ Wuv,
                                                 float* __restrict__ attn_out) {
  __shared__ float cx[CKV_];
  const int b = blockIdx.y, n = blockIdx.x, d = threadIdx.x;
  for (int i = threadIdx.x; i < CKV_; i += 128) cx[i] = ctx[((size_t)b * NH_ + n) * CKV_ + i];
  __syncthreads();
  float acc = 0.f;
#pragma unroll 4
  for (int vv = 0; vv < CKV_; ++vv)
    acc = fmaf(cx[vv], Wuv[((size_t)vv * NH_ + n) * HD_ + d], acc);
  attn_out[((size_t)b * NH_ + n) * HD_ + d] = acc;
}

// ---------------------------------------------------------------------------
// Cache concatenation (grid-stride float4 copies).
// ---------------------------------------------------------------------------
__global__ void concat_ckv_k(const float* __restrict__ cache, const float* __restrict__ nrow,
                             float* __restrict__ out) {
  const size_t total4 = (size_t)B_ * Tp1 * (CKV_ / 4);
  for (size_t i = (size_t)blockIdx.x * blockDim.x + threadIdx.x; i < total4;
       i += (size_t)gridDim.x * blockDim.x) {
    const size_t b = i / ((size_t)Tp1 * (CKV_ / 4));
    const size_t rem = i - b * (size_t)Tp1 * (CKV_ / 4);
    const size_t t = rem / (CKV_ / 4);
    const size_t c = rem % (CKV_ / 4);
    float4 val = (t < T_) ? ((const float4*)cache)[((size_t)b * T_ + t) * (CKV_ / 4) + c]
                          : ((const float4*)nrow)[b * (CKV_ / 4) + c];
    ((float4*)out)[i] = val;
  }
}

__global__ void concat_kr_k(const float* __restrict__ cache, const float* __restrict__ nrow,
                            float* __restrict__ out) {
  const int ROW4 = NH_ * RD_ / 4;  // 512
  const size_t total4 = (size_t)B_ * Tp1 * ROW4;
  for (size_t i = (size_t)blockIdx.x * blockDim.x + threadIdx.x; i < total4;
       i += (size_t)gridDim.x * blockDim.x) {
    const size_t b = i / ((size_t)Tp1 * ROW4);
    const size_t rem = i - b * (size_t)Tp1 * ROW4;
    const size_t t = rem / ROW4;
    const size_t c = rem % ROW4;
    float4 val = (t < T_) ? ((const float4*)cache)[((size_t)b * T_ + t) * ROW4 + c]
                          : ((const float4*)nrow)[b * ROW4 + c];
    ((float4*)out)[i] = val;
  }
}

// ---------------------------------------------------------------------------
extern "C" void kernel_launch(void* const* d_in, const int* in_sizes, int n_in,
                              void* d_out, int out_size, void* d_ws, size_t ws_size,
                              hipStream_t stream) {
  const float* hidden = (const float*)d_in[0];   // [B,1,HID]
  const float* mask   = (const float*)d_in[1];   // [B,1,NH,T+1]
  const float* c_ckv  = (const float*)d_in[2];   // [B,T,CKV]
  const float* c_kr   = (const float*)d_in[3];   // [B,T,NH,RD]
  const float* W_DQ   = (const float*)d_in[4];   // [HID,CQ]
  const float* W_DKV  = (const float*)d_in[5];   // [HID,CKV]
  const float* W_UQC  = (const float*)d_in[6];   // [CQ,NH,HD]
  const float* W_UQR  = (const float*)d_in[7];   // [CQ,NH,RD]
  const float* W_UKC  = (const float*)d_in[8];   // [CKV,NH,HD]
  const float* W_UVC  = (const float*)d_in[9];   // [CKV,NH,HD]
  const float* W_KR   = (const float*)d_in[10];  // [HID,NH,RD]
  const float* W_O    = (const float*)d_in[11];  // [NH,HD,HID]

  float* out_y   = (float*)d_out;                         // [B,1,HID]
  float* out_ckv = out_y + (size_t)B_ * HID;              // [B,T+1,CKV]
  float* out_kr  = out_ckv + (size_t)B_ * Tp1 * CKV_;     // [B,T+1,NH,RD]

  float* ws = (float*)d_ws;
  float* cQ     = ws;                 ws += (size_t)B_ * CQ_;
  float* cKVt   = ws;                 ws += (size_t)B_ * CKV_;
  float* kRt    = ws;                 ws += (size_t)B_ * NH_ * RD_;
  float* qC     = ws;                 ws += (size_t)B_ * NH_ * HD_;
  float* qR     = ws;                 ws += (size_t)B_ * NH_ * RD_;
  float* kCt    = ws;                 ws += (size_t)B_ * NH_ * HD_;
  float* q_abs  = ws;                 ws += (size_t)B_ * NH_ * CKV_;
  float* scores = ws;                 ws += (size_t)B_ * NH_ * Tp1;
  float* ctx    = ws;                 ws += (size_t)B_ * NH_ * CKV_;
  float* a_out  = ws;                 ws += (size_t)B_ * NH_ * HD_;

  // --- projections (RoPE at position 0 is the identity; skipped) ---
  gemv_k<<<dim3(CQ_ / 256, B_), 256, HID * 4, stream>>>(hidden, W_DQ, cQ, HID, CQ_);
  gemv_k<<<dim3(CKV_ / 256, B_), 256, HID * 4, stream>>>(hidden, W_DKV, cKVt, HID, CKV_);
  gemv_k<<<dim3(NH_ * RD_ / 256, B_), 256, HID * 4, stream>>>(hidden, W_KR, kRt, HID, NH_ * RD_);
  gemv_k<<<dim3(NH_ * HD_ / 256, B_), 256, CQ_ * 4, stream>>>(cQ, W_UQC, qC, CQ_, NH_ * HD_);
  gemv_k<<<dim3(NH_ * RD_ / 256, B_), 256, CQ_ * 4, stream>>>(cQ, W_UQR, qR, CQ_, NH_ * RD_);
  gemv_k<<<dim3(NH_ * HD_ / 256, B_), 256, CKV_ * 4, stream>>>(cKVt, W_UKC, kCt, CKV_, NH_ * HD_);

  // --- absorb W_UK_C into q ---
  qabs_k<<<dim3(NH_, B_), 256, 0, stream>>>(qC, W_UKC, q_abs);

  // --- attention scores ---
  scores_wmma_k<<<dim3(T_ / 128, B_), 256, 0, stream>>>(c_ckv, q_abs, scores);
  rope_score_k<<<dim3(T_ / 8, B_), 256, 0, stream>>>(c_kr, qR, mask, scores);
  new_score_k<<<1, B_ * NH_, 0, stream>>>(qC, kCt, qR, kRt, mask, scores);
  softmax_k<<<dim3(NH_, B_), 256, 0, stream>>>(scores);

  // --- aggregation (absorbed V) + output projection ---
  agg_wmma_k<<<dim3(CKV_ / 128, B_), 256, 0, stream>>>(scores, c_ckv, cKVt, ctx);
  attnout_k<<<dim3(NH_, B_), 128, 0, stream>>>(ctx, W_UVC, a_out);
  gemv_k<<<dim3(HID / 256, B_), 256, NH_ * HD_ * 4, stream>>>(a_out, W_O, out_y, NH_ * HD_, HID);

  // --- cache concatenation ---
  concat_ckv_k<<<2048, 256, 0, stream>>>(c_ckv, cKVt, out_ckv);
  concat_kr_k<<<4096, 256, 0, stream>>>(c_kr, kRt, out_kr);
}